// HieraTripletLoss_71030169141308
// MI455X (gfx1250) — compile-verified
//
#include <hip/hip_runtime.h>

#define N_FINE   19
#define N_COARSE 4
#define NCH      23
#define EPSV     1e-8f
#define MULTV    5.0f
#define IGNORE_L 255

#define BATCH   4
#define HH      512
#define WW      1024
#define HW      (HH * WW)            /* 524288 = 2^19 */
#define PIXELS  (BATCH * HW)         /* 2097152 */
#define THREADS 256
#define BLOCKS  1024
#define ITERS   (PIXELS / (THREADS * BLOCKS))   /* 8, exact */

typedef float v2f __attribute__((ext_vector_type(2)));
typedef float v8f __attribute__((ext_vector_type(8)));

__device__ __forceinline__ int cmapf(int f) {
    return (f < 2) ? 0 : ((f < 8) ? 1 : ((f < 13) ? 2 : 3));
}

// log(sigmoid(z) + eps)
__device__ __forceinline__ float logsigeps(float z) {
    float e = __expf(-z);
    float s = __builtin_amdgcn_rcpf(1.0f + e);
    return __logf(s + EPSV);
}

// Wave32 sum via V_WMMA_F32_16X16X4_F32 with B = ones.
// A VGPR0 = v (lanes 0-15 -> A[m][0], lanes 16-31 -> A[m][2]), A VGPR1 = 0.
// D[m][n] = v_m + v_{16+m} for every column n. Lane L (0-15) holds rows 0-7
// of column L, lane L+16 holds rows 8-15 -> sum own 8 regs + shfl from +16.
__device__ __forceinline__ float wave_sum32(float v) {
    v2f a; a[0] = v;    a[1] = 0.0f;
    v2f b; b[0] = 1.0f; b[1] = 1.0f;
    v8f c = {};
    v8f d = __builtin_amdgcn_wmma_f32_16x16x4_f32(false, a, false, b,
                                                  (short)0, c, false, false);
    float s = d[0] + d[1] + d[2] + d[3] + d[4] + d[5] + d[6] + d[7];
    s += __shfl_down(s, 16, 32);
    return s;   // full wave sum valid in lanes 0..15
}

__global__ __launch_bounds__(THREADS) void hiera_zero5(float* g) {
    if (threadIdx.x < 5) g[threadIdx.x] = 0.0f;
}

__global__ __launch_bounds__(THREADS) void hiera_main(const float* __restrict__ cls,
                                                      const int*   __restrict__ label,
                                                      float*       __restrict__ gacc) {
    __shared__ float sacc[5];
    if (threadIdx.x < 5) sacc[threadIdx.x] = 0.0f;
    __syncthreads();

    float aF = 0.f, aC = 0.f, aCEf = 0.f, aCEc = 0.f, aV = 0.f;
    const int tid0 = blockIdx.x * THREADS + threadIdx.x;

    for (int it = 0; it < ITERS; ++it) {
        const int pix = tid0 + it * (THREADS * BLOCKS);
        const int b   = pix >> 19;          // HW = 2^19
        const int hw  = pix & (HW - 1);
        const float* p = cls + (size_t)b * (size_t)(NCH * HW) + (size_t)hw;

        float xs[NCH];
#pragma unroll
        for (int c = 0; c < NCH; ++c) xs[c] = p[(size_t)c * HW];

        const int t = label[pix];
        if (t != IGNORE_L) {
            const int tf = t;
            const int tc = cmapf(tf);
            aV += 1.0f;

            // pass 1: group maxes over fine logits, global fine max, gather x[tf]
            float g4[4] = {-3.4e38f, -3.4e38f, -3.4e38f, -3.4e38f};
            float mxf = -3.4e38f, xtf = 0.f;
#pragma unroll
            for (int f = 0; f < N_FINE; ++f) {
                const float xf = xs[f];
                mxf = fmaxf(mxf, xf);
                xtf = (f == tf) ? xf : xtf;
                g4[cmapf(f)] = fmaxf(g4[cmapf(f)], xf);   // constant index (unrolled)
            }

            // masked CE, fine: lse - x[tf]
            float se = 0.f;
#pragma unroll
            for (int f = 0; f < N_FINE; ++f) se += __expf(xs[f] - mxf);
            aCEf += mxf + __logf(se) - xtf;

            // masked CE, coarse
            float xctc = 0.f;
            const float mxc = fmaxf(fmaxf(xs[19], xs[20]), fmaxf(xs[21], xs[22]));
            float sec = 0.f;
#pragma unroll
            for (int i = 0; i < N_COARSE; ++i) {
                xctc = (i == tc) ? xs[19 + i] : xctc;
                sec += __expf(xs[19 + i] - mxc);
            }
            aCEc += mxc + __logf(sec) - xctc;

            // hiera fine: pos -> -log(sigmoid(min(xf, xc))+eps)
            //             neg -> -log(1-sigmoid(xf)+eps) = -log(sigmoid(-xf)+eps)
#pragma unroll
            for (int f = 0; f < N_FINE; ++f) {
                const float xcm = xs[19 + cmapf(f)];
                const float arg = (f == tf) ? fminf(xs[f], xcm) : (-xs[f]);
                aF -= logsigeps(arg);
            }

            // hiera coarse: MCMB logit = max(group_max, xc_i)
#pragma unroll
            for (int i = 0; i < N_COARSE; ++i) {
                const float mb  = fmaxf(g4[i], xs[19 + i]);
                const float arg = (i == tc) ? xs[19 + i] : (-mb);
                aC -= logsigeps(arg);
            }
        }
    }

    // All 256 threads reach here together -> EXEC all-1s for WMMA.
    const float wF = wave_sum32(aF);
    const float wC = wave_sum32(aC);
    const float w2 = wave_sum32(aCEf);
    const float w3 = wave_sum32(aCEc);
    const float w4 = wave_sum32(aV);

    if ((threadIdx.x & 31) == 0) {
        atomicAdd(&sacc[0], wF);
        atomicAdd(&sacc[1], wC);
        atomicAdd(&sacc[2], w2);
        atomicAdd(&sacc[3], w3);
        atomicAdd(&sacc[4], w4);
    }
    __syncthreads();
    if (threadIdx.x < 5) atomicAdd(&gacc[threadIdx.x], sacc[threadIdx.x]);
}

__global__ void hiera_finalize(const float* __restrict__ g, float* __restrict__ out) {
    const float V  = fmaxf(g[4], 1.0f);
    const float lf = g[0] / (V * (float)N_FINE);
    const float lc = g[1] / (V * (float)N_COARSE);
    out[0] = MULTV * (lf + lc) + g[2] / V + g[3] / V;
}

extern "C" void kernel_launch(void* const* d_in, const int* in_sizes, int n_in,
                              void* d_out, int out_size, void* d_ws, size_t ws_size,
                              hipStream_t stream) {
    (void)in_sizes; (void)n_in; (void)out_size; (void)ws_size;
    const float* cls   = (const float*)d_in[0];   // (4, 23, 512, 1024) f32
    const int*   label = (const int*)d_in[2];     // (4, 512, 1024) i32
    float*       gacc  = (float*)d_ws;            // 5 f32 accumulators
    float*       outp  = (float*)d_out;           // scalar f32

    hiera_zero5<<<1, 32, 0, stream>>>(gacc);
    hiera_main<<<BLOCKS, THREADS, 0, stream>>>(cls, label, gacc);
    hiera_finalize<<<1, 1, 0, stream>>>(gacc, outp);
}